// GCN_11587821765342
// MI455X (gfx1250) — compile-verified
//
#include <hip/hip_runtime.h>

typedef __attribute__((ext_vector_type(2))) float v2f;
typedef __attribute__((ext_vector_type(8))) float v8f;

// ---------------- degree / norm ----------------
__global__ void deg_init_kernel(float* deg, int n) {
    int i = blockIdx.x * blockDim.x + threadIdx.x;
    if (i < n) deg[i] = 1.0f;  // self-loop contributes 1 to every node's degree
}

__global__ void deg_accum_kernel(const long long* __restrict__ dst, float* deg, int E) {
    int e = blockIdx.x * blockDim.x + threadIdx.x;
    if (e < E) atomicAdd(&deg[(int)dst[e]], 1.0f);
}

__global__ void deg_rsqrt_kernel(float* deg, int n) {
    int i = blockIdx.x * blockDim.x + threadIdx.x;
    if (i < n) {
        float d = deg[i];
        deg[i] = d > 0.0f ? rsqrtf(d) : 0.0f;
    }
}

// Zero-pad W2 [64 x 40] into W2p [64 x 48] so the GEMM needs no load guards.
__global__ void pad_w2_kernel(const float* __restrict__ W2, float* __restrict__ W2p) {
    int i = blockIdx.x * blockDim.x + threadIdx.x;
    if (i >= 64 * 48) return;
    int r = i / 48, c = i % 48;
    W2p[i] = (c < 40) ? W2[r * 40 + c] : 0.0f;
}

// ---- fp32 WMMA GEMM with fused aggregation-seed epilogue ----
// T[n x FOUT] = op(H)[n x K] @ W[K x FW]   (op = ReLU if RELU_IN)
// O[n x FOUT] = bias + T * dinv^2          (bias + self-loop seed for edge scatter)
// FW multiple of 16 (weights pre-padded); FOUT <= FW is store width/stride.
// One wave per 16x16 tile, V_WMMA_F32_16X16X4_F32, branchless inner loop.
template <int K, int FW, int FOUT, bool RELU_IN>
__global__ void gemm_wmma_kernel(const float* __restrict__ H, const float* __restrict__ W,
                                 float* __restrict__ T, const float* __restrict__ dinv,
                                 const float* __restrict__ bias, float* __restrict__ O,
                                 int n) {
    static_assert(FW % 16 == 0, "padded weight width must be a multiple of 16");
    constexpr int FT = FW / 16;
    int wid    = blockIdx.x * (blockDim.x >> 5) + (threadIdx.x >> 5);
    int mtiles = (n + 15) >> 4;
    if (wid >= mtiles * FT) return;                 // wave-uniform: EXEC stays all-1s

    int mt = wid / FT, ft = wid % FT;
    int row0 = mt << 4, col0 = ft << 4;
    int lane = threadIdx.x & 31;
    int l15  = lane & 15;
    int koff = (lane >> 4) << 1;                    // lanes 16-31 hold K+2,K+3

    int arow = row0 + l15; if (arow > n - 1) arow = n - 1;   // branchless clamp
    int bcol = col0 + l15;                          // always < FW (padded)
    const float* __restrict__ ap = H + (size_t)arow * K + koff;
    const float* __restrict__ wp = W + (size_t)koff * FW + bcol;

    v8f acc = {};
#pragma unroll 4
    for (int kb = 0; kb < K; kb += 4) {
        v2f a, b;
        float a0 = ap[kb], a1 = ap[kb + 1];
        if constexpr (RELU_IN) { a0 = fmaxf(a0, 0.0f); a1 = fmaxf(a1, 0.0f); }
        a.x = a0;
        a.y = a1;
        b.x = wp[(size_t)kb * FW];
        b.y = wp[(size_t)(kb + 1) * FW];
        // (neg_a, A, neg_b, B, c_mod, C, reuse_a, reuse_b)
        acc = __builtin_amdgcn_wmma_f32_16x16x4_f32(false, a, false, b, (short)0, acc,
                                                    false, false);
    }

    // ---- fused epilogue: store T and bias+self-loop seed O ----
    int  mbase = row0 + ((lane >> 4) << 3);         // C/D rows: M = r + 8*(lane>=16)
    int  bc    = (FW == FOUT) ? bcol : (bcol < FOUT ? bcol : FOUT - 1);
    float bv   = bias[bc];
    bool colok = (FW == FOUT) || (bcol < FOUT);     // compile-time true for 64-wide layers

    if (colok) {                                    // post-WMMA divergence: harmless
        float*       tp = T + (size_t)mbase * FOUT + bcol;
        float*       op = O + (size_t)mbase * FOUT + bcol;
        const float* dp = dinv + mbase;
        if (row0 + 16 <= n) {                       // wave-uniform fast path (N%16==0)
#pragma unroll
            for (int r = 0; r < 8; ++r) {
                float t  = acc[r];
                float di = dp[r];
                tp[(size_t)r * FOUT] = t;
                op[(size_t)r * FOUT] = bv + t * di * di;
            }
        } else {
#pragma unroll
            for (int r = 0; r < 8; ++r) {
                if (mbase + r < n) {
                    float t  = acc[r];
                    float di = dp[r];
                    tp[(size_t)r * FOUT] = t;
                    op[(size_t)r * FOUT] = bv + t * di * di;
                }
            }
        }
    }
}

// For each edge: out[dst] += T[src] * dinv[src]*dinv[dst]  (float4 gather, f32 atomics)
template <int F>
__global__ void agg_edges_kernel(const float* __restrict__ T,
                                 const long long* __restrict__ src,
                                 const long long* __restrict__ dst,
                                 const float* __restrict__ dinv,
                                 float* __restrict__ out, int E) {
    constexpr int CH  = (F + 3) / 4;   // float4 chunks per row (16 for F=64, 10 for F=40)
    constexpr int LPE = 16;            // lanes per edge
    long long gid = (long long)blockIdx.x * blockDim.x + threadIdx.x;
    int e = (int)(gid >> 4);
    int c = (int)(gid & (LPE - 1));
    if (e >= E || c >= CH) return;
    int s = (int)src[e];
    int d = (int)dst[e];
    float nrm = dinv[s] * dinv[d];
    const float4 v = *(const float4*)(T + (size_t)s * F + (size_t)c * 4);
    float* o = out + (size_t)d * F + (size_t)c * 4;
    atomicAdd(o + 0, v.x * nrm);
    atomicAdd(o + 1, v.y * nrm);
    atomicAdd(o + 2, v.z * nrm);
    atomicAdd(o + 3, v.w * nrm);
}

// ---------------- orchestration ----------------
extern "C" void kernel_launch(void* const* d_in, const int* in_sizes, int n_in,
                              void* d_out, int out_size, void* d_ws, size_t ws_size,
                              hipStream_t stream) {
    const float*     x    = (const float*)d_in[0];
    const long long* ei   = (const long long*)d_in[1];
    const float*     W0   = (const float*)d_in[2];
    const float*     b0   = (const float*)d_in[3];
    const float*     W1   = (const float*)d_in[4];
    const float*     b1   = (const float*)d_in[5];
    const float*     W2   = (const float*)d_in[6];
    const float*     b2   = (const float*)d_in[7];
    float*           outp = (float*)d_out;

    const int N = in_sizes[0] / 128;
    const int E = in_sizes[1] / 2;
    const long long* srcp = ei;        // edge_index[0]
    const long long* dstp = ei + E;    // edge_index[1]

    float* dinv  = (float*)d_ws;
    size_t off   = ((size_t)N + 255) & ~(size_t)255;
    float* bufT  = dinv + off;                 // GEMM output (messages to gather)
    float* bufH  = bufT + (size_t)N * 64;      // aggregated features
    float* W2pad = bufH + (size_t)N * 64;      // 64 x 48 zero-padded W2

    const int TB = 256;
    // --- degrees / norm + weight padding (independent, front-loaded) ---
    deg_init_kernel<<<(N + TB - 1) / TB, TB, 0, stream>>>(dinv, N);
    pad_w2_kernel<<<(64 * 48 + TB - 1) / TB, TB, 0, stream>>>(W2, W2pad);
    deg_accum_kernel<<<(E + TB - 1) / TB, TB, 0, stream>>>(dstp, dinv, E);
    deg_rsqrt_kernel<<<(N + TB - 1) / TB, TB, 0, stream>>>(dinv, N);

    auto gemm_blocks = [&](int FW) {
        int tiles = ((N + 15) / 16) * (FW / 16);
        return (tiles + 7) / 8;   // 8 waves (16x16 tiles) per 256-thread block
    };
    int edge_blocks = (int)(((long long)E * 16 + TB - 1) / TB);

    // --- layer 0: T = x@W0, O seeded with b0 + self-loop, then edge scatter ---
    gemm_wmma_kernel<128, 64, 64, false><<<gemm_blocks(64), TB, 0, stream>>>(
        x, W0, bufT, dinv, b0, bufH, N);
    agg_edges_kernel<64><<<edge_blocks, TB, 0, stream>>>(bufT, srcp, dstp, dinv, bufH, E);

    // --- layer 1: ReLU fused into A-loads; T = relu(h1)@W1 ---
    gemm_wmma_kernel<64, 64, 64, true><<<gemm_blocks(64), TB, 0, stream>>>(
        bufH, W1, bufT, dinv, b1, bufH, N);
    agg_edges_kernel<64><<<edge_blocks, TB, 0, stream>>>(bufT, srcp, dstp, dinv, bufH, E);

    // --- layer 2: 64 -> 40 (weights padded to 48), aggregate straight into d_out ---
    gemm_wmma_kernel<64, 48, 40, true><<<gemm_blocks(48), TB, 0, stream>>>(
        bufH, W2pad, bufT, dinv, b2, outp, N);
    agg_edges_kernel<40><<<edge_blocks, TB, 0, stream>>>(bufT, srcp, dstp, dinv, outp, E);
}